// SamplingLayer_22660247453989
// MI455X (gfx1250) — compile-verified
//
#include <hip/hip_runtime.h>
#include <hip/hip_bf16.h>

#define D_EV 2
#define H 256
#define NUM_FLOWS 2

typedef float v2f __attribute__((ext_vector_type(2)));
typedef float v8f __attribute__((ext_vector_type(8)));

// ---------------------------------------------------------------------------
// Precompute per-flow affine coefficients of the collapsed MAF:
//   y0 = x0 * exp(b3[1]) + b3[0]
//   y1 = x1 * exp(A_l*y0 + B_l) + A_s*y0 + B_s
// where A_*/B_* come from W1row0 / b1 pushed through W2 and W3 cols 2,3.
// Q[n][j] = sum_k W2[j,k] * W3[k, 2+n]   (computed with V_WMMA_F32_16X16X4_F32)
// coef layout per flow f (6 floats): { b3[0], exp(b3[1]), A_s, B_s, A_l, B_l }
// ---------------------------------------------------------------------------
__global__ __launch_bounds__(512)
void maf_precompute(const float* __restrict__ W1, const float* __restrict__ b1,
                    const float* __restrict__ W2, const float* __restrict__ b2,
                    const float* __restrict__ W3, const float* __restrict__ b3,
                    float* __restrict__ coef) {
  const int f    = blockIdx.x;          // flow index
  const int tid  = threadIdx.x;         // 0..511  (16 waves of 32)
  const int wave = tid >> 5;            // row-tile index m = 0..15
  const int lane = tid & 31;
  const int lh   = lane >> 4;           // lane half
  const int ll   = lane & 15;

  const float* W2f = W2 + (size_t)f * H * H;
  const float* W3f = W3 + (size_t)f * H * (2 * D_EV);

  __shared__ float Q[2][H];             // Q[n][j]
  __shared__ float red[6][H];           // block reduction scratch

#if __has_builtin(__builtin_amdgcn_wmma_f32_16x16x4_f32)
  {
    // One 16x16 output tile per wave: rows 16*m .. 16*m+15, cols 0..15
    // (only cols 0,1 carry real data; B cols >=2 are zero).
    v8f acc = {};
    const int m = wave;
    const float* arow = W2f + (size_t)(16 * m + ll) * H;   // A row for this lane
    for (int kk = 0; kk < 64; ++kk) {
      const int kbase = 4 * kk + 2 * lh;                    // K of A/B vgpr0
      v2f a, b;
      a.x = arow[kbase + 0];                                // A[m_row][kbase]
      a.y = arow[kbase + 1];                                // A[m_row][kbase+1]
      if (ll < 2) {                                         // B[k][n] = W3f[k*4 + 2+n]
        b.x = W3f[(kbase + 0) * 4 + 2 + ll];
        b.y = W3f[(kbase + 1) * 4 + 2 + ll];
      } else {
        b.x = 0.0f; b.y = 0.0f;
      }
      acc = __builtin_amdgcn_wmma_f32_16x16x4_f32(
          /*neg_a=*/false, a, /*neg_b=*/false, b,
          /*c_mod=*/(short)0, acc, /*reuse_a=*/false, /*reuse_b=*/false);
    }
    // D layout: vgpr r, lane-half lh -> M = r + 8*lh, N = ll
    if (ll < 2) {
#pragma unroll
      for (int r = 0; r < 8; ++r)
        Q[ll][16 * m + 8 * lh + r] = acc[r];
    }
  }
#else
  // Scalar fallback (no WMMA): 512 threads -> 512 outputs of Q[2][256]
  {
    const int n = tid & 1;
    const int j = tid >> 1;
    float s = 0.0f;
    const float* row = W2f + (size_t)j * H;
    for (int k = 0; k < H; ++k) s += row[k] * W3f[k * 4 + 2 + n];
    Q[n][j] = s;
  }
#endif
  __syncthreads();

  // Phase 2: six length-256 dot products, block tree reduction.
  if (tid < H) {
    const float w1  = W1[(size_t)f * D_EV * H + tid];  // W1[f][0][tid]
    const float bb1 = b1[(size_t)f * H + tid];
    const float bb2 = b2[(size_t)f * H + tid];
    const float q0  = Q[0][tid], q1 = Q[1][tid];
    const float w32 = W3f[tid * 4 + 2];
    const float w33 = W3f[tid * 4 + 3];
    red[0][tid] = w1  * q0;   // -> A_s
    red[1][tid] = w1  * q1;   // -> A_l
    red[2][tid] = bb1 * q0;   // -> part of B_s
    red[3][tid] = bb1 * q1;   // -> part of B_l
    red[4][tid] = bb2 * w32;  // -> part of B_s
    red[5][tid] = bb2 * w33;  // -> part of B_l
  }
  __syncthreads();
  for (int s = H / 2; s > 0; s >>= 1) {
    if (tid < s) {
#pragma unroll
      for (int i = 0; i < 6; ++i) red[i][tid] += red[i][tid + s];
    }
    __syncthreads();
  }
  if (tid == 0) {
    const float* b3f = b3 + (size_t)f * (2 * D_EV);
    float* cf = coef + f * 6;
    cf[0] = b3f[0];                        // shift0 (constant)
    cf[1] = __expf(b3f[1]);                // exp(log_scale0) (constant)
    cf[2] = red[0][0];                     // A_s
    cf[3] = red[2][0] + red[4][0] + b3f[2];// B_s
    cf[4] = red[1][0];                     // A_l
    cf[5] = red[3][0] + red[5][0] + b3f[3];// B_l
  }
}

// ---------------------------------------------------------------------------
// Main elementwise kernel: reparameterization + 2 collapsed MAF flows.
// float4 path: each thread handles 2 samples (2x float2). HBM-bound:
// 16 MB total traffic -> ~0.7 us at 23.3 TB/s.
// ---------------------------------------------------------------------------
__global__ __launch_bounds__(256)
void maf_apply(const float4* __restrict__ zm, const float4* __restrict__ zlv,
               const float4* __restrict__ eps, const float* __restrict__ coef,
               float4* __restrict__ out, int nPairs) {
  const int i = blockIdx.x * blockDim.x + threadIdx.x;
  if (i >= nPairs) return;

  float c[12];
#pragma unroll
  for (int k = 0; k < 12; ++k) c[k] = coef[k];   // uniform -> scalar loads

  const float4 m4 = zm[i];
  const float4 v4 = zlv[i];
  const float4 e4 = eps[i];

  float z0a = m4.x + __expf(0.5f * v4.x) * e4.x;
  float z1a = m4.y + __expf(0.5f * v4.y) * e4.y;
  float z0b = m4.z + __expf(0.5f * v4.z) * e4.z;
  float z1b = m4.w + __expf(0.5f * v4.w) * e4.w;

#pragma unroll
  for (int f = 0; f < NUM_FLOWS; ++f) {
    const float* cf = c + 6 * f;
    const float y0a = z0a * cf[1] + cf[0];
    const float y1a = z1a * __expf(cf[4] * y0a + cf[5]) + cf[2] * y0a + cf[3];
    const float y0b = z0b * cf[1] + cf[0];
    const float y1b = z1b * __expf(cf[4] * y0b + cf[5]) + cf[2] * y0b + cf[3];
    z0a = y0a; z1a = y1a;
    z0b = y0b; z1b = y1b;
  }

  float4 o; o.x = z0a; o.y = z1a; o.z = z0b; o.w = z1b;
  out[i] = o;
}

extern "C" void kernel_launch(void* const* d_in, const int* in_sizes, int n_in,
                              void* d_out, int out_size, void* d_ws, size_t ws_size,
                              hipStream_t stream) {
  const float* zm  = (const float*)d_in[0];
  const float* zlv = (const float*)d_in[1];
  const float* eps = (const float*)d_in[2];
  const float* W1  = (const float*)d_in[3];
  const float* b1  = (const float*)d_in[4];
  const float* W2  = (const float*)d_in[5];
  const float* b2  = (const float*)d_in[6];
  const float* W3  = (const float*)d_in[7];
  const float* b3  = (const float*)d_in[8];

  float* coef = (float*)d_ws;   // 12 floats

  maf_precompute<<<NUM_FLOWS, 512, 0, stream>>>(W1, b1, W2, b2, W3, b3, coef);

  const int nPairs = in_sizes[0] / 4;   // (B*D)/4 = 2 samples per thread
  const int block = 256;
  const int grid = (nPairs + block - 1) / block;
  maf_apply<<<grid, block, 0, stream>>>((const float4*)zm, (const float4*)zlv,
                                        (const float4*)eps, coef,
                                        (float4*)d_out, nPairs);
}